// MSAFBlock_22866405884027
// MI455X (gfx1250) — compile-verified
//
#include <hip/hip_runtime.h>

// ---------------------------------------------------------------------------
// MSAF block for MI455X (gfx1250).
// Bandwidth-bound: 603 MB minimum HBM traffic, reduced to ~412 MB by reading
// Phase-C inputs in reverse-b order (192 MB L2 holds the tail of the 201 MB
// input stream) and streaming outputs with non-temporal stores.
// The two tiny f32 GEMMs use V_WMMA_F32_16X16X4_F32 (full fp32 precision).
// ---------------------------------------------------------------------------

typedef float v2f __attribute__((ext_vector_type(2)));
typedef float v4f __attribute__((ext_vector_type(4)));
typedef float v8f __attribute__((ext_vector_type(8)));

#define BC   256      // block channels
#define RC   64       // reduced channels
#define NB0  4
#define NB1  2
#define KTOT 6        // NB0 + NB1
#define B_   16       // batch
#define C0_  1024
#define C1_  512
#define L_   2048
#define EPS  1e-5f

// ---------------------------------------------------------------------------
// Kernel 1: gap[b, c] = mean_L( sum_blocks x0 + sum_blocks x1 )
// One workgroup per (b, c): sums 6 rows of 2048 floats with b128 loads,
// then a fixed-order LDS tree reduction (deterministic, no float atomics).
// ---------------------------------------------------------------------------
__global__ __launch_bounds__(256) void msaf_gap_kernel(
    const float* __restrict__ x0, const float* __restrict__ x1,
    float* __restrict__ gap)
{
    __shared__ float red[256];
    const int b = blockIdx.x >> 8;
    const int c = blockIdx.x & 255;
    const int t = threadIdx.x;

    float s = 0.f;
#pragma unroll
    for (int n = 0; n < NB0; ++n) {
        const v4f* row = (const v4f*)(x0 + (size_t)(b * C0_ + n * BC + c) * L_);
        v4f u = row[t];
        v4f v = row[t + 256];
        s += (u.x + u.y + u.z + u.w) + (v.x + v.y + v.z + v.w);
    }
#pragma unroll
    for (int n = 0; n < NB1; ++n) {
        const v4f* row = (const v4f*)(x1 + (size_t)(b * C1_ + n * BC + c) * L_);
        v4f u = row[t];
        v4f v = row[t + 256];
        s += (u.x + u.y + u.z + u.w) + (v.x + v.y + v.z + v.w);
    }

    red[t] = s;
    __syncthreads();
    for (int off = 128; off > 0; off >>= 1) {
        if (t < off) red[t] += red[t + off];
        __syncthreads();
    }
    if (t == 0) gap[blockIdx.x] = red[0] * (1.0f / (float)L_);
}

// ---------------------------------------------------------------------------
// Kernel 2: full attention path with V_WMMA_F32_16X16X4_F32.
//   Stage 1 (waves 0..3): h[16,64] = ReLU(BN(gap @ Wj^T + bj)), N-tiled by 16.
//   Stage 2 (waves 0..15): per c-tile, 6 accumulators (one per block) held in
//   registers; softmax over the 6 blocks is elementwise across accumulators.
// A-frag f32 16x4: lanes 0-15 hold K=k,k+1 ; lanes 16-31 hold K=k+2,k+3.
// C/D tile: VGPR j -> M = j (+8 for lanes 16-31), N = lane & 15.
// ---------------------------------------------------------------------------
__global__ __launch_bounds__(512) void msaf_atten_kernel(
    const float* __restrict__ gap,   // [16,256]
    const float* __restrict__ Wj,    // [64,256]
    const float* __restrict__ bj,    // [64]
    const float* __restrict__ gamma, const float* __restrict__ beta,
    const float* __restrict__ mean,  const float* __restrict__ var,
    const float* __restrict__ Wg,    // [6,256,64]
    const float* __restrict__ bg,    // [6,256]
    float* __restrict__ atten)       // [6,16,256]
{
    __shared__ float h_lds[B_][RC];          // 4 KB

    const int lane = threadIdx.x & 31;
    const int wave = threadIdx.x >> 5;
    const int half = lane >> 4;              // 0: lanes 0-15, 1: lanes 16-31
    const int l16  = lane & 15;

    // ---- Stage 1: joint linear + BN + ReLU -> h_lds ----
    if (wave < 4) {
        const int rt = wave * 16;            // output-column (r) tile base
        const int r  = rt + l16;             // this lane's B column
        v8f acc = {};
        for (int k = 0; k < BC; k += 4) {
            const int ka = k + half * 2;
            v2f a, bf;
            a.x  = gap[l16 * BC + ka];
            a.y  = gap[l16 * BC + ka + 1];
            bf.x = Wj[r * BC + ka];          // B[k][n] = Wj[n][k]
            bf.y = Wj[r * BC + ka + 1];
            acc = __builtin_amdgcn_wmma_f32_16x16x4_f32(
                      false, a, false, bf, (short)0, acc, false, false);
        }
        const float sc = rsqrtf(var[r] + EPS) * gamma[r];
        const float sh = (bj[r] - mean[r]) * sc + beta[r];
#pragma unroll
        for (int j = 0; j < 8; ++j) {
            const int m = j + half * 8;
            float hv = acc[j] * sc + sh;
            h_lds[m][r] = hv > 0.f ? hv : 0.f;
        }
    }
    __syncthreads();

    // ---- Stage 2: grouped linears + softmax over blocks, one c-tile/wave ----
    {
        const int c0 = wave * 16;            // 16 waves x 16 cols = 256
        const int c  = c0 + l16;
        v8f acc[KTOT] = {};
        for (int r = 0; r < RC; r += 4) {
            const int ra = r + half * 2;
            v2f a;
            a.x = h_lds[l16][ra];
            a.y = h_lds[l16][ra + 1];
#pragma unroll
            for (int k = 0; k < KTOT; ++k) {
                const float* wk = Wg + (size_t)(k * BC + c) * RC; // B[r][c]=Wg[k,c,r]
                v2f bf;
                bf.x = wk[ra];
                bf.y = wk[ra + 1];
                acc[k] = __builtin_amdgcn_wmma_f32_16x16x4_f32(
                             false, a, false, bf, (short)0, acc[k], false, false);
            }
        }
        float bias[KTOT];
#pragma unroll
        for (int k = 0; k < KTOT; ++k) bias[k] = bg[k * BC + c];

#pragma unroll
        for (int j = 0; j < 8; ++j) {
            const int brow = j + half * 8;   // batch index of this element
            float lg[KTOT];
            float mx = -3.402823466e38f;
#pragma unroll
            for (int k = 0; k < KTOT; ++k) {
                lg[k] = acc[k][j] + bias[k];
                mx = fmaxf(mx, lg[k]);
            }
            float sum = 0.f;
#pragma unroll
            for (int k = 0; k < KTOT; ++k) {
                lg[k] = expf(lg[k] - mx);
                sum += lg[k];
            }
            const float inv = 1.0f / sum;
#pragma unroll
            for (int k = 0; k < KTOT; ++k)
                atten[(size_t)(k * B_ + brow) * BC + c] = lg[k] * inv;
        }
    }
}

// ---------------------------------------------------------------------------
// Kernel 3: out = x * atten[block]. One workgroup per (b, channel) row.
// Reverse-b traversal so Phase-A's L2-resident tail is read first (~95% L2
// hits out of 192 MB); outputs streamed with non-temporal b128 stores so they
// don't evict the inputs we still need.
// ---------------------------------------------------------------------------
__global__ __launch_bounds__(256) void msaf_scale_kernel(
    const float* __restrict__ x0, const float* __restrict__ x1,
    const float* __restrict__ atten, float* __restrict__ out)
{
    const int gr = (int)blockIdx.x;          // 0 .. 16*1536-1
    const int b  = (B_ - 1) - gr / (C0_ + C1_);
    const int r  = gr % (C0_ + C1_);

    const float* src;
    float* dst;
    int k, c;
    if (r < C0_) {                           // modality 0
        k = r >> 8;  c = r & 255;
        const size_t off = (size_t)(b * C0_ + r) * L_;
        src = x0 + off;
        dst = out + off;
    } else {                                 // modality 1
        const int rr = r - C0_;
        k = NB0 + (rr >> 8);  c = rr & 255;
        const size_t off = (size_t)(b * C1_ + rr) * L_;
        src = x1 + off;
        dst = out + (size_t)B_ * C0_ * L_ + off;
    }

    const float s = atten[(size_t)(k * B_ + b) * BC + c];
    const v4f* sp = (const v4f*)src;
    v4f* dp = (v4f*)dst;
    const int t = threadIdx.x;

    v4f u = sp[t];
    v4f v = sp[t + 256];
    u *= s;
    v *= s;
    __builtin_nontemporal_store(u, dp + t);
    __builtin_nontemporal_store(v, dp + t + 256);
}

// ---------------------------------------------------------------------------
extern "C" void kernel_launch(void* const* d_in, const int* in_sizes, int n_in,
                              void* d_out, int out_size, void* d_ws, size_t ws_size,
                              hipStream_t stream) {
    (void)in_sizes; (void)n_in; (void)out_size; (void)ws_size;

    const float* x0    = (const float*)d_in[0];
    const float* x1    = (const float*)d_in[1];
    const float* Wj    = (const float*)d_in[2];
    const float* bj    = (const float*)d_in[3];
    const float* gamma = (const float*)d_in[4];
    const float* beta  = (const float*)d_in[5];
    const float* mean  = (const float*)d_in[6];
    const float* var   = (const float*)d_in[7];
    const float* Wg    = (const float*)d_in[8];
    const float* bg    = (const float*)d_in[9];
    float* out = (float*)d_out;

    // workspace: gap [16*256] then atten [6*16*256]  (~115 KB total)
    float* gap   = (float*)d_ws;
    float* atten = gap + B_ * BC;

    msaf_gap_kernel<<<B_ * BC, 256, 0, stream>>>(x0, x1, gap);
    msaf_atten_kernel<<<1, 512, 0, stream>>>(gap, Wj, bj, gamma, beta, mean, var,
                                             Wg, bg, atten);
    msaf_scale_kernel<<<B_ * (C0_ + C1_), 256, 0, stream>>>(x0, x1, atten, out);
}